// CopyMechanism_15530601742393
// MI455X (gfx1250) — compile-verified
//
#include <hip/hip_runtime.h>
#include <hip/hip_bf16.h>
#include <cstdint>

#define B_    8
#define T_    256
#define V_    32000
#define E_    512
#define OOV_  100
#define VE_   (V_ + OOV_)      // 32100
#define BT_   (B_ * T_)        // 2048

typedef __attribute__((ext_vector_type(2))) float v2f;
typedef __attribute__((ext_vector_type(8))) float v8f;

// ---------------------------------------------------------------------------
// Kernel A: pgen = sigmoid(concat(ctx,dhs,din) . w + b) via V_WMMA_F32_16X16X4_F32
// One wave handles 16 consecutive (b,t) rows. A-layout (32-bit A 16x4):
// lanes 0-15 hold K=k,k+1 ; lanes 16-31 hold K=k+2,k+3 for row M=lane&15.
// B broadcasts w[k] across all 16 columns -> every D column equals the dot.
// ---------------------------------------------------------------------------
__device__ __forceinline__ void wmma_seg(v8f& c, const float* __restrict__ rowp,
                                         const float* __restrict__ wp,
                                         int kn, int koff) {
#pragma unroll 4
  for (int k = 0; k < kn; k += 4) {
    v2f a = *(const v2f*)(rowp + k + koff);
    v2f b = *(const v2f*)(wp + k + koff);
    c = __builtin_amdgcn_wmma_f32_16x16x4_f32(false, a, false, b,
                                              (short)0, c, false, false);
  }
}

__global__ __launch_bounds__(32) void pgen_wmma_kernel(
    const float* __restrict__ ctx,   // (BT,512)
    const float* __restrict__ dhs,   // (BT,1024)
    const float* __restrict__ din,   // (BT,512)
    const float* __restrict__ w,     // (2048)
    const float* __restrict__ bias,  // (1)
    float* __restrict__ pgen_out)    // (BT)
{
  const int lane = threadIdx.x & 31;
  const int half = lane >> 4;        // 0: K pair {k,k+1}, 1: {k+2,k+3}
  const int m    = lane & 15;        // matrix row within tile
  const int row  = blockIdx.x * 16 + m;
  const int koff = 2 * half;

  v8f c = {};
  wmma_seg(c, ctx + (size_t)row * 512,  w,         512, koff);
  wmma_seg(c, dhs + (size_t)row * 1024, w + 512,  1024, koff);
  wmma_seg(c, din + (size_t)row * 512,  w + 1536,  512, koff);

  const float bb = bias[0];
  // C/D layout: VGPR r -> M=r (lanes 0-15) / M=r+8 (lanes 16-31); column = lane&15.
  // All columns identical, so lane 0 emits rows 0..7 and lane 16 emits rows 8..15.
  if (m == 0) {
    const int base = blockIdx.x * 16 + (half ? 8 : 0);
#pragma unroll
    for (int r = 0; r < 8; ++r) {
      float s = c[r] + bb;
      pgen_out[base + r] = 1.0f / (1.0f + __expf(-s));
    }
  }
}

// ---------------------------------------------------------------------------
// Kernel B: per-row softmax over V=32000, write total = pgen * probs_ext
// (OOV tail written as 0). One 1024-thread block per (b,t) row; 32 regs/thread.
// Logits streamed once with non-temporal loads.
// ---------------------------------------------------------------------------
#define EPT 32   // 1024*32 = 32768 >= 32000; only j==31 is partial (tid<256)

__global__ __launch_bounds__(1024) void softmax_scale_kernel(
    const float* __restrict__ logits,   // (BT, V)
    const float* __restrict__ pgen_arr, // (BT)
    float* __restrict__ out)            // (BT, VE)
{
  const int row = blockIdx.x;
  const int tid = threadIdx.x;
  const float* lrow = logits + (size_t)row * V_;

  float x[EPT];
  float m = -3.4e38f;
#pragma unroll
  for (int j = 0; j < EPT; ++j) {
    const int v = tid + j * 1024;
    if (j < EPT - 1 || tid < (V_ - (EPT - 1) * 1024)) {
      x[j] = __builtin_nontemporal_load(lrow + v);
    } else {
      x[j] = -3.4e38f;
    }
    m = fmaxf(m, x[j]);
  }

  __shared__ float sred[32];
  const int wid = tid >> 5, lane = tid & 31;

  // block max
#pragma unroll
  for (int off = 16; off > 0; off >>= 1) m = fmaxf(m, __shfl_xor(m, off, 32));
  if (lane == 0) sred[wid] = m;
  __syncthreads();
  if (wid == 0) {
    float v = sred[lane];
#pragma unroll
    for (int off = 16; off > 0; off >>= 1) v = fmaxf(v, __shfl_xor(v, off, 32));
    if (lane == 0) sred[0] = v;
  }
  __syncthreads();
  const float M = sred[0];
  __syncthreads();   // everyone has read M before sred is reused

  float s = 0.0f;
#pragma unroll
  for (int j = 0; j < EPT; ++j) {
    x[j] = __expf(x[j] - M);   // padded lanes: exp(-huge) -> 0
    s += x[j];
  }

  // block sum
#pragma unroll
  for (int off = 16; off > 0; off >>= 1) s += __shfl_xor(s, off, 32);
  if (lane == 0) sred[wid] = s;
  __syncthreads();
  if (wid == 0) {
    float v = sred[lane];
#pragma unroll
    for (int off = 16; off > 0; off >>= 1) v += __shfl_xor(v, off, 32);
    if (lane == 0) sred[0] = v;
  }
  __syncthreads();

  const float pg    = pgen_arr[row];
  const float scale = pg / sred[0];
  float* orow = out + (size_t)row * VE_;
#pragma unroll
  for (int j = 0; j < EPT; ++j) {
    const int v = tid + j * 1024;
    if (j < EPT - 1 || tid < (V_ - (EPT - 1) * 1024))
      orow[v] = x[j] * scale;
  }
  if (tid < OOV_) orow[V_ + tid] = 0.0f;  // pgen * 0 pad
}

// ---------------------------------------------------------------------------
// Kernel C: total[b,t, enc[b,e]] += (1-pgen[b,t]) * attn[b,t,e]
// Row-local global atomics; rows were just written so mostly L2-resident.
// ---------------------------------------------------------------------------
__global__ __launch_bounds__(256) void scatter_copy_kernel(
    const float* __restrict__ attn,     // (BT, E)
    const int*   __restrict__ enc,      // (B, E)
    const float* __restrict__ pgen_arr, // (BT)
    float* __restrict__ out)            // (BT, VE)
{
  const int i  = blockIdx.x * 256 + threadIdx.x;  // < BT_*E_
  const int bt = i >> 9;                          // / E_
  const int e  = i & (E_ - 1);
  const int b  = bt >> 8;                         // / T_
  const float pc  = 1.0f - pgen_arr[bt];
  const int   idx = enc[b * E_ + e];
  atomicAdd(out + (size_t)bt * VE_ + idx, pc * attn[i]);
}

// ---------------------------------------------------------------------------
extern "C" void kernel_launch(void* const* d_in, const int* in_sizes, int n_in,
                              void* d_out, int out_size, void* d_ws, size_t ws_size,
                              hipStream_t stream) {
  const float* logits = (const float*)d_in[0];  // (B,T,V)
  const float* attn   = (const float*)d_in[1];  // (B,T,E)
  const float* dhs    = (const float*)d_in[2];  // (B,T,1024)
  const float* din    = (const float*)d_in[3];  // (B,T,512)
  const float* ctx    = (const float*)d_in[4];  // (B,T,512)
  const int*   enc    = (const int*)d_in[5];    // (B,E)
  const float* w      = (const float*)d_in[6];  // (1,2048)
  const float* bias   = (const float*)d_in[7];  // (1,)
  // d_in[8] = max_oovs (compile-time constant 100 here)

  float* out  = (float*)d_out;                       // total: BT*VE floats
  float* pgen = out + (size_t)BT_ * VE_;             // pgen tail: BT floats

  pgen_wmma_kernel<<<BT_ / 16, 32, 0, stream>>>(ctx, dhs, din, w, bias, pgen);
  softmax_scale_kernel<<<BT_, 1024, 0, stream>>>(logits, pgen, out);
  scatter_copy_kernel<<<(BT_ * E_) / 256, 256, 0, stream>>>(attn, enc, pgen, out);
}